// GCN_55284819034805
// MI455X (gfx1250) — compile-verified
//
#include <hip/hip_runtime.h>

typedef __attribute__((ext_vector_type(2))) float v2f;
typedef __attribute__((ext_vector_type(8))) float v8f;

// ---------------------------------------------------------------------------
// Kernel 1: init workspace. deg = 1.0 (self loop), agg1 = 0, agg2 = 0.
// Launched with >= 16*N threads.
// ---------------------------------------------------------------------------
__global__ void gcn_init(float* __restrict__ deg, float* __restrict__ agg1,
                         float* __restrict__ agg2, int n) {
    int i = blockIdx.x * blockDim.x + threadIdx.x;
    if (i < n) { deg[i] = 1.0f; agg2[i] = 0.0f; }
    if (i < n * 16) { agg1[i] = 0.0f; }
}

// ---------------------------------------------------------------------------
// Kernel 2: degree accumulation over edge targets (self loops pre-seeded).
// ---------------------------------------------------------------------------
__global__ void gcn_degree(const int* __restrict__ col, float* __restrict__ deg,
                           int E) {
    int e = blockIdx.x * blockDim.x + threadIdx.x;
    if (e >= E) return;
    atomicAdd(&deg[col[e]], 1.0f);
}

// ---------------------------------------------------------------------------
// Kernel 3: dinv = rsqrt(deg)   (deg >= 1 always, matches reference's where)
// ---------------------------------------------------------------------------
__global__ void gcn_dinv(const float* __restrict__ deg, float* __restrict__ dinv,
                         int n) {
    int i = blockIdx.x * blockDim.x + threadIdx.x;
    if (i >= n) return;
    float d = deg[i];
    dinv[i] = (d > 0.0f) ? rsqrtf(d) : 0.0f;
}

// ---------------------------------------------------------------------------
// Kernel 4: h1 = X @ W1 in exact f32 using V_WMMA_F32_16X16X4_F32.
//   X : [n,128] row-major, W1 : [128,16] row-major, h1 : [n,16] row-major.
// One wave per 16-row tile; 32 K-steps of the 16x16x4 f32 WMMA.
//   A (16x4): lane l<16 holds row M=l, K={0,1}; lane l+16 holds K={2,3}.
//   B (4x16): lane l<16 holds col N=l, K={0,1}; lane l+16 holds K={2,3}.
//   C (16x16): VGPR r, lanes 0-15 -> M=r, lanes 16-31 -> M=r+8; N = lane&15.
// ---------------------------------------------------------------------------
__global__ void gcn_gemm1_wmma(const float* __restrict__ x,
                               const float* __restrict__ W1,
                               float* __restrict__ h1, int ntiles) {
    int wave = blockIdx.x * (blockDim.x >> 5) + (threadIdx.x >> 5);
    if (wave >= ntiles) return;              // wave-uniform: EXEC stays all-ones
    int lane  = threadIdx.x & 31;
    int m     = lane & 15;                   // row within tile / col of B
    int kb    = (lane >> 4) << 1;            // 0 or 2
    size_t row16 = (size_t)wave * 16;

    v8f acc = {};
    const float* arow = x + (row16 + (size_t)m) * 128 + kb;
    #pragma unroll 4
    for (int k0 = 0; k0 < 128; k0 += 4) {
        v2f a = *(const v2f*)(arow + k0);                 // x[row, k0+kb .. +1]
        v2f b;
        b.x = W1[(size_t)(k0 + kb) * 16 + m];             // W1[k0+kb  , m]
        b.y = W1[(size_t)(k0 + kb + 1) * 16 + m];         // W1[k0+kb+1, m]
        acc = __builtin_amdgcn_wmma_f32_16x16x4_f32(
            /*neg_a=*/false, a, /*neg_b=*/false, b,
            /*c_mod=*/(short)0, acc, /*reuse_a=*/false, /*reuse_b=*/false);
    }

    int mbase = (lane >> 4) << 3;            // 0 or 8
    int ncol  = lane & 15;
    #pragma unroll
    for (int r = 0; r < 8; ++r) {
        h1[(row16 + (size_t)(mbase + r)) * 16 + ncol] = acc[r];
    }
}

// Scalar tail for n % 16 != 0 (not hit for N=200000, kept for generality).
__global__ void gcn_gemm1_tail(const float* __restrict__ x,
                               const float* __restrict__ W1,
                               float* __restrict__ h1, int start, int n) {
    int t = blockIdx.x * blockDim.x + threadIdx.x;
    int node = start + (t >> 4);
    int f = t & 15;
    if (node >= n) return;
    float acc = 0.0f;
    const float* xr = x + (size_t)node * 128;
    for (int k = 0; k < 128; ++k) acc += xr[k] * W1[(size_t)k * 16 + f];
    h1[(size_t)node * 16 + f] = acc;
}

// ---------------------------------------------------------------------------
// Kernel 5: edge aggregation, 16 features. 16 lanes per edge.
//   agg1[col, f] += dinv[row]*dinv[col] * h1[row, f]
// ---------------------------------------------------------------------------
__global__ void gcn_edge16(const int* __restrict__ row, const int* __restrict__ col,
                           const float* __restrict__ dinv,
                           const float* __restrict__ h1,
                           float* __restrict__ agg1, int E) {
    int t = blockIdx.x * blockDim.x + threadIdx.x;
    int e = t >> 4;
    int f = t & 15;
    if (e >= E) return;
    int r = row[e];
    int c = col[e];
    float nrm = dinv[r] * dinv[c];
    float v = nrm * h1[(size_t)r * 16 + f];
    atomicAdd(&agg1[(size_t)c * 16 + f], v);
}

// ---------------------------------------------------------------------------
// Kernel 6: post layer-1 (adds self-loop + bias, relu) fused with h @ W2.
//   s[i] = sum_f relu(agg1[i,f] + dinv[i]^2*h1[i,f] + b1[f]) * W2[f]
// ---------------------------------------------------------------------------
__global__ void gcn_post1(const float* __restrict__ agg1,
                          const float* __restrict__ h1,
                          const float* __restrict__ dinv,
                          const float* __restrict__ b1,
                          const float* __restrict__ W2,
                          float* __restrict__ s, int n) {
    int i = blockIdx.x * blockDim.x + threadIdx.x;
    if (i >= n) return;
    float di = dinv[i];
    float self = di * di;
    const float* a = agg1 + (size_t)i * 16;
    const float* h = h1 + (size_t)i * 16;
    float acc = 0.0f;
    #pragma unroll
    for (int f = 0; f < 16; ++f) {
        float v = a[f] + self * h[f] + b1[f];
        v = fmaxf(v, 0.0f);
        acc = fmaf(v, W2[f], acc);
    }
    s[i] = acc;
}

// ---------------------------------------------------------------------------
// Kernel 7: scalar edge aggregation for layer 2.
//   agg2[col] += dinv[row]*dinv[col] * s[row]
// ---------------------------------------------------------------------------
__global__ void gcn_edge1(const int* __restrict__ row, const int* __restrict__ col,
                          const float* __restrict__ dinv,
                          const float* __restrict__ s,
                          float* __restrict__ agg2, int E) {
    int e = blockIdx.x * blockDim.x + threadIdx.x;
    if (e >= E) return;
    int r = row[e];
    int c = col[e];
    atomicAdd(&agg2[c], dinv[r] * dinv[c] * s[r]);
}

// ---------------------------------------------------------------------------
// Kernel 8: final output.  out[i] = agg2[i] + dinv[i]^2 * s[i] + b2
// ---------------------------------------------------------------------------
__global__ void gcn_post2(const float* __restrict__ agg2,
                          const float* __restrict__ s,
                          const float* __restrict__ dinv,
                          const float* __restrict__ b2,
                          float* __restrict__ out, int n) {
    int i = blockIdx.x * blockDim.x + threadIdx.x;
    if (i >= n) return;
    float di = dinv[i];
    out[i] = agg2[i] + di * di * s[i] + b2[0];
}

// ---------------------------------------------------------------------------
extern "C" void kernel_launch(void* const* d_in, const int* in_sizes, int n_in,
                              void* d_out, int out_size, void* d_ws, size_t ws_size,
                              hipStream_t stream) {
    const float* x  = (const float*)d_in[0];   // [N,128]
    const int*   ei = (const int*)d_in[1];     // [2,E]
    const float* W1 = (const float*)d_in[2];   // [128,16]
    const float* b1 = (const float*)d_in[3];   // [16]
    const float* W2 = (const float*)d_in[4];   // [16,1]
    const float* b2 = (const float*)d_in[5];   // [1]

    const int N = in_sizes[0] / 128;
    const int E = in_sizes[1] / 2;
    const int* row = ei;        // sources j
    const int* col = ei + E;    // targets i

    // Workspace layout (floats): deg[N] dinv[N] h1[16N] agg1[16N] s[N] agg2[N]
    float* ws   = (float*)d_ws;
    float* deg  = ws;
    float* dinv = deg + N;
    float* h1   = dinv + N;
    float* agg1 = h1 + (size_t)N * 16;
    float* s    = agg1 + (size_t)N * 16;
    float* agg2 = s + N;
    float* out  = (float*)d_out;

    const int B = 256;

    // 1. init (needs 16N threads)
    {
        int total = N * 16;
        gcn_init<<<(total + B - 1) / B, B, 0, stream>>>(deg, agg1, agg2, N);
    }
    // 2. degree
    gcn_degree<<<(E + B - 1) / B, B, 0, stream>>>(col, deg, E);
    // 3. dinv
    gcn_dinv<<<(N + B - 1) / B, B, 0, stream>>>(deg, dinv, N);
    // 4. h1 = X @ W1 (WMMA f32)
    {
        int full   = N & ~15;
        int ntiles = full >> 4;
        if (ntiles > 0) {
            int wavesPerBlock = B / 32;
            int blocks = (ntiles + wavesPerBlock - 1) / wavesPerBlock;
            gcn_gemm1_wmma<<<blocks, B, 0, stream>>>(x, W1, h1, ntiles);
        }
        if (full < N) {
            int tail = (N - full) * 16;
            gcn_gemm1_tail<<<(tail + B - 1) / B, B, 0, stream>>>(x, W1, h1, full, N);
        }
    }
    // 5. layer-1 edge aggregation (16 lanes per edge)
    {
        long long total = (long long)E * 16;
        gcn_edge16<<<(int)((total + B - 1) / B), B, 0, stream>>>(row, col, dinv, h1, agg1, E);
    }
    // 6. relu + bias + self-loop + dot(W2)
    gcn_post1<<<(N + B - 1) / B, B, 0, stream>>>(agg1, h1, dinv, b1, W2, s, N);
    // 7. layer-2 edge aggregation (scalar)
    gcn_edge1<<<(E + B - 1) / B, B, 0, stream>>>(row, col, dinv, s, agg2, E);
    // 8. final
    gcn_post2<<<(N + B - 1) / B, B, 0, stream>>>(agg2, s, dinv, b2, out, N);
}